// PointNetFeaturePropagation_24575802868371
// MI455X (gfx1250) — compile-verified
//
#include <hip/hip_runtime.h>
#include <hip/hip_bf16.h>

// ---------------------------------------------------------------------------
// PointNetFeaturePropagation for MI455X (gfx1250, wave32, WMMA)
//   B=4, N=8192, S=2048, D=256, C_in=512, mlp=[512,256]
// ---------------------------------------------------------------------------

typedef float v2f __attribute__((ext_vector_type(2)));
typedef float v8f __attribute__((ext_vector_type(8)));

#define BN_EPS 1e-3f

#define Bb   4
#define Nn   8192
#define Ss   2048
#define Dd   256
#define CIN  512      // 2*D
#define H0   512      // mlp[0]
#define H1   256      // mlp[1]
#define Mtot (Bb * Nn)

// ---------------------------------------------------------------------------
// Kernel 1: 3-NN search + inverse-distance weights.
// One thread per query point n; xyz2[b] (3*2048 floats = 24KB) staged in LDS.
// ---------------------------------------------------------------------------
__global__ __launch_bounds__(256) void knn_kernel(
    const float* __restrict__ xyz1, const float* __restrict__ xyz2,
    int* __restrict__ idx_out, float* __restrict__ w_out)
{
    __shared__ float lx[Ss], ly[Ss], lz[Ss];
    const int b = blockIdx.x / (Nn / 256);
    const int n = (blockIdx.x % (Nn / 256)) * 256 + threadIdx.x;

    const float* x2 = xyz2 + (size_t)b * 3 * Ss;
    for (int s = threadIdx.x; s < Ss; s += 256) {
        lx[s] = x2[s];
        ly[s] = x2[Ss + s];
        lz[s] = x2[2 * Ss + s];
    }
    __syncthreads();

    const float* x1 = xyz1 + (size_t)b * 3 * Nn;
    const float px = x1[n], py = x1[Nn + n], pz = x1[2 * Nn + n];

    float d0 = 3.4e38f, d1 = 3.4e38f, d2 = 3.4e38f;
    int   i0 = 0, i1 = 0, i2 = 0;
    for (int s = 0; s < Ss; ++s) {
        const float dx = px - lx[s];
        const float dy = py - ly[s];
        const float dz = pz - lz[s];
        const float dd = fmaf(dx, dx, fmaf(dy, dy, dz * dz));
        if (dd < d0)      { d2 = d1; i2 = i1; d1 = d0; i1 = i0; d0 = dd; i0 = s; }
        else if (dd < d1) { d2 = d1; i2 = i1; d1 = dd; i1 = s; }
        else if (dd < d2) { d2 = dd; i2 = s; }
    }
    d0 = fmaxf(d0, 1e-10f); d1 = fmaxf(d1, 1e-10f); d2 = fmaxf(d2, 1e-10f);
    float w0 = 1.f / d0, w1 = 1.f / d1, w2 = 1.f / d2;
    const float inv = 1.f / (w0 + w1 + w2);

    const size_t m = (size_t)b * Nn + n;
    idx_out[m * 3 + 0] = i0; idx_out[m * 3 + 1] = i1; idx_out[m * 3 + 2] = i2;
    w_out[m * 3 + 0] = w0 * inv; w_out[m * 3 + 1] = w1 * inv; w_out[m * 3 + 2] = w2 * inv;
}

// ---------------------------------------------------------------------------
// Kernel 2: gather 3 neighbor features + concat with points1 -> X[Mtot, 512].
// points2 (8.4MB) fits in the 192MB L2, so the gather scatter stays on-chip.
// One block per point, one thread per channel d (coalesced X writes).
// ---------------------------------------------------------------------------
__global__ __launch_bounds__(256) void buildx_kernel(
    const float* __restrict__ points1, const float* __restrict__ points2,
    const int* __restrict__ idx, const float* __restrict__ w,
    float* __restrict__ X)
{
    const int m = blockIdx.x;       // 0..Mtot-1
    const int d = threadIdx.x;      // 0..255
    const int b = m / Nn, n = m % Nn;

    const int   i0 = idx[m * 3 + 0], i1 = idx[m * 3 + 1], i2 = idx[m * 3 + 2];
    const float w0 = w[m * 3 + 0],  w1 = w[m * 3 + 1],  w2 = w[m * 3 + 2];

    const float* p2 = points2 + (size_t)b * Dd * Ss + (size_t)d * Ss;
    const float interp = fmaf(w0, p2[i0], fmaf(w1, p2[i1], w2 * p2[i2]));
    const float p1v = points1[(size_t)b * Dd * Nn + (size_t)d * Nn + n];

    X[(size_t)m * CIN + d]      = p1v;
    X[(size_t)m * CIN + Dd + d] = interp;
}

// ---------------------------------------------------------------------------
// Kernel 3/4: GEMM (M x 512) * (512 x NOUT) + bias + inference-BN + ReLU on
// the fp32 WMMA path (v_wmma_f32_16x16x4_f32).
//
// Register blocking: block = 256 threads = 8 wave32s in a 4(M) x 2(N) grid;
// each wave owns a 32x32 C tile (four v8f accumulators). Per k-step a wave
// loads 2 A fragments + 2 B fragments (4 x ds_read_b64) and issues 4 WMMAs:
// 1 LDS fragment load per WMMA (vs 2 in the naive 16x16 version).
//
// Block tile 128(M) x 64(N), Kc = 32 staged in LDS (As 17KB + Bs 8.7KB).
// Fragment layout per ISA 7.12.2 (32-bit A 16x4): lane&15 = M (N for B),
// lane>>4 selects the K pair; Bs is stored [n][k] so both fragment loads are
// 8B-aligned ds_read_b64 (rows padded to 34 floats, conflict-free).
//
// TOUT=true writes Out[b, n, m%N] (the final transpose) with contiguous
// per-lane float4 stores (each lane owns one output channel).
// ---------------------------------------------------------------------------
template <int NOUT, bool TOUT>
__global__ __launch_bounds__(256) void gemm_bn_relu(
    const float* __restrict__ A, const float* __restrict__ W,
    const float* __restrict__ bias, const float* __restrict__ gamma,
    const float* __restrict__ beta, const float* __restrict__ mean,
    const float* __restrict__ var, float* __restrict__ Out)
{
    constexpr int K  = 512;
    constexpr int KC = 32;

    __shared__ __align__(16) float As[128][34];
    __shared__ __align__(16) float Bs[64][34];   // [n][k]

    const int ntiles = NOUT / 64;
    const int tm = blockIdx.x / ntiles;
    const int tn = blockIdx.x % ntiles;
    const int row0 = tm * 128;
    const int col0 = tn * 64;

    const int t    = threadIdx.x;
    const int lane = t & 31;
    const int wave = t >> 5;
    const int wm   = wave & 3;         // 0..3  (M sub-tile, 32 rows each)
    const int wn   = wave >> 2;        // 0..1  (N sub-tile, 32 cols each)
    const int lm   = lane & 15;        // fragment M (A) / N (B) index
    const int koff = (lane >> 4) << 1; // K pair select: 0 or 2

    v8f c00 = {}, c01 = {}, c10 = {}, c11 = {};

    for (int k0 = 0; k0 < K; k0 += KC) {
        // ---- stage A tile (128 x 32): 16 contiguous floats per thread ----
        {
            const int r  = t >> 1;               // 0..127
            const int cg = (t & 1) << 4;         // 0 or 16
            const float4* src =
                (const float4*)(A + (size_t)(row0 + r) * K + k0 + cg);
            const float4 v0 = src[0];
            const float4 v1 = src[1];
            const float4 v2 = src[2];
            const float4 v3 = src[3];
            *(float4*)&As[r][cg]      = v0;
            *(float4*)&As[r][cg + 4]  = v1;
            *(float4*)&As[r][cg + 8]  = v2;
            *(float4*)&As[r][cg + 12] = v3;
        }
        // ---- stage B tile (32 x 64), transposed into Bs[n][k] ----
        {
            const int kk = t >> 3;               // 0..31
            const int cg = (t & 7) << 3;         // 0,8,...,56
            const float4 v0 =
                *(const float4*)(W + (size_t)(k0 + kk) * NOUT + col0 + cg);
            const float4 v1 =
                *(const float4*)(W + (size_t)(k0 + kk) * NOUT + col0 + cg + 4);
            Bs[cg + 0][kk] = v0.x;
            Bs[cg + 1][kk] = v0.y;
            Bs[cg + 2][kk] = v0.z;
            Bs[cg + 3][kk] = v0.w;
            Bs[cg + 4][kk] = v1.x;
            Bs[cg + 5][kk] = v1.y;
            Bs[cg + 6][kk] = v1.z;
            Bs[cg + 7][kk] = v1.w;
        }
        __syncthreads();

#pragma unroll
        for (int kk = 0; kk < KC; kk += 4) {
            const v2f a0 = *(const v2f*)&As[wm * 32 + lm][kk + koff];
            const v2f a1 = *(const v2f*)&As[wm * 32 + 16 + lm][kk + koff];
            const v2f b0 = *(const v2f*)&Bs[wn * 32 + lm][kk + koff];
            const v2f b1 = *(const v2f*)&Bs[wn * 32 + 16 + lm][kk + koff];
            c00 = __builtin_amdgcn_wmma_f32_16x16x4_f32(
                      false, a0, false, b0, (short)0, c00, false, false);
            c01 = __builtin_amdgcn_wmma_f32_16x16x4_f32(
                      false, a0, false, b1, (short)0, c01, false, false);
            c10 = __builtin_amdgcn_wmma_f32_16x16x4_f32(
                      false, a1, false, b0, (short)0, c10, false, false);
            c11 = __builtin_amdgcn_wmma_f32_16x16x4_f32(
                      false, a1, false, b1, (short)0, c11, false, false);
        }
        __syncthreads();
    }

    // ---- fused epilogue: bias + inference BN + ReLU ----
    const int hi = lane >> 4;

#pragma unroll
    for (int ni = 0; ni < 2; ++ni) {
        const int   n  = col0 + wn * 32 + ni * 16 + lm;   // output channel
        const float sc = gamma[n] * rsqrtf(var[n] + BN_EPS);
        const float sh = beta[n] - mean[n] * sc;
        const float bz = bias[n];

#pragma unroll
        for (int mi = 0; mi < 2; ++mi) {
            const v8f cc   = (mi == 0) ? (ni == 0 ? c00 : c01)
                                       : (ni == 0 ? c10 : c11);
            const int mbase = row0 + wm * 32 + mi * 16 + hi * 8;

            if constexpr (!TOUT) {
                // Row-major Out[m][NOUT]; lanes 0..15 hit consecutive n.
#pragma unroll
                for (int r = 0; r < 8; ++r) {
                    const float y = fmaf(cc[r] + bz, sc, sh);
                    Out[(size_t)(mbase + r) * NOUT + n] = fmaxf(y, 0.f);
                }
            } else {
                // Out[b, n, m % Nn]: per-lane m contiguous -> 2 float4 stores.
                const int bidx = mbase / Nn;   // 128-row tiles never span batches
                const int np   = mbase % Nn;
                float* dst = Out + (size_t)bidx * NOUT * Nn + (size_t)n * Nn + np;
                float4 o0, o1;
                o0.x = fmaxf(fmaf(cc[0] + bz, sc, sh), 0.f);
                o0.y = fmaxf(fmaf(cc[1] + bz, sc, sh), 0.f);
                o0.z = fmaxf(fmaf(cc[2] + bz, sc, sh), 0.f);
                o0.w = fmaxf(fmaf(cc[3] + bz, sc, sh), 0.f);
                o1.x = fmaxf(fmaf(cc[4] + bz, sc, sh), 0.f);
                o1.y = fmaxf(fmaf(cc[5] + bz, sc, sh), 0.f);
                o1.z = fmaxf(fmaf(cc[6] + bz, sc, sh), 0.f);
                o1.w = fmaxf(fmaf(cc[7] + bz, sc, sh), 0.f);
                *(float4*)dst       = o0;
                *(float4*)(dst + 4) = o1;
            }
        }
    }
}

// ---------------------------------------------------------------------------
// Host launcher
// ---------------------------------------------------------------------------
extern "C" void kernel_launch(void* const* d_in, const int* in_sizes, int n_in,
                              void* d_out, int out_size, void* d_ws, size_t ws_size,
                              hipStream_t stream)
{
    const float* xyz1    = (const float*)d_in[0];
    const float* xyz2    = (const float*)d_in[1];
    const float* points1 = (const float*)d_in[2];
    const float* points2 = (const float*)d_in[3];
    const float* w0 = (const float*)d_in[4];
    const float* b0 = (const float*)d_in[5];
    const float* g0 = (const float*)d_in[6];
    const float* be0 = (const float*)d_in[7];
    const float* m0 = (const float*)d_in[8];
    const float* v0 = (const float*)d_in[9];
    const float* w1 = (const float*)d_in[10];
    const float* b1 = (const float*)d_in[11];
    const float* g1 = (const float*)d_in[12];
    const float* be1 = (const float*)d_in[13];
    const float* m1 = (const float*)d_in[14];
    const float* v1 = (const float*)d_in[15];

    float* out = (float*)d_out;

    // Workspace layout: X (64MB) | h (64MB) | idx (384KB) | w (384KB)
    float* X   = (float*)d_ws;
    float* h   = X + (size_t)Mtot * CIN;
    int*   idx = (int*)(h + (size_t)Mtot * H0);
    float* wgt = (float*)(idx + (size_t)Mtot * 3);

    // 1) 3-NN + weights
    knn_kernel<<<Bb * (Nn / 256), 256, 0, stream>>>(xyz1, xyz2, idx, wgt);

    // 2) gather + concat -> X[Mtot, 512]
    buildx_kernel<<<Mtot, 256, 0, stream>>>(points1, points2, idx, wgt, X);

    // 3) layer 0: h = relu(bn(X @ w0 + b0))   [Mtot, 512]
    gemm_bn_relu<H0, false><<<(Mtot / 128) * (H0 / 64), 256, 0, stream>>>(
        X, w0, b0, g0, be0, m0, v0, h);

    // 4) layer 1: out = relu(bn(h @ w1 + b1)) transposed to [B, 256, N]
    gemm_bn_relu<H1, true><<<(Mtot / 128) * (H1 / 64), 256, 0, stream>>>(
        h, w1, b1, g1, be1, m1, v1, out);
}